// Qwen3NextTopKRouter_32392643347143
// MI455X (gfx1250) — compile-verified
//
#include <hip/hip_runtime.h>
#include <hip/hip_bf16.h>

typedef __attribute__((ext_vector_type(2))) float v2f;
typedef __attribute__((ext_vector_type(4))) float v4f;
typedef __attribute__((ext_vector_type(8))) float v8f;

#define HDIM 2048
#define NEXP 64
#define TOPK 8
#define ROWS_PER_WAVE 32          // two 16-row A tiles share the four B tiles
#define WAVES_PER_BLOCK 4
#define ROWS_PER_BLOCK (ROWS_PER_WAVE * WAVES_PER_BLOCK)   // 128

// ---------------------------------------------------------------------------
// GEMM: logits[T,64] = hs[T,2048] x wt[64,2048]^T using V_WMMA_F32_16X16X4_F32
// Each wave: 32 rows x 64 cols = eight 16x16 f32 C tiles (64 acc VGPRs).
// Per k-step of 4: 2 A-frag b64 loads + 4 B-frag b64 loads feed 8 WMMAs
// (B frags reused across both row tiles -> 8 WMMA : 6 loads issue ratio).
// A frag (16x4 f32): lane l holds row (base + l%16), K pair {k+2h, k+2h+1},
// h = l/16 -> contiguous 8B load. B frag (4x16 f32): lane l holds col l%16,
// same K pair -> contiguous 8B load from the weight row (wt is [E,H] row-major).
// ---------------------------------------------------------------------------
__global__ __launch_bounds__(128)
void router_gemm_wmma(const float* __restrict__ hs,
                      const float* __restrict__ wt,
                      float* __restrict__ logits)
{
    const int lane = threadIdx.x & 31;
    const int wave = threadIdx.x >> 5;
    const int idx  = lane & 15;     // row index (A) / col index (B) within tile
    const int half = lane >> 4;     // selects K pair {2h, 2h+1}
    const int rowBase = blockIdx.x * ROWS_PER_BLOCK + wave * ROWS_PER_WAVE;

    const float* aptr0 = hs + (size_t)(rowBase +  0 + idx) * HDIM + 2 * half;
    const float* aptr1 = hs + (size_t)(rowBase + 16 + idx) * HDIM + 2 * half;
    const float* b0    = wt + (size_t)( 0 + idx) * HDIM + 2 * half;
    const float* b1    = wt + (size_t)(16 + idx) * HDIM + 2 * half;
    const float* b2    = wt + (size_t)(32 + idx) * HDIM + 2 * half;
    const float* b3    = wt + (size_t)(48 + idx) * HDIM + 2 * half;

    v8f acc0a = {}, acc1a = {}, acc2a = {}, acc3a = {};
    v8f acc0b = {}, acc1b = {}, acc2b = {}, acc3b = {};

#pragma unroll 2
    for (int k = 0; k < HDIM; k += 4) {
        // hidden_states is streamed exactly once (134 MB) -> non-temporal
        v2f aA = __builtin_nontemporal_load((const v2f*)(aptr0 + k));
        v2f aB = __builtin_nontemporal_load((const v2f*)(aptr1 + k));
        // weight is 512 KB, reused by every row tile -> cacheable
        v2f f0 = *(const v2f*)(b0 + k);
        v2f f1 = *(const v2f*)(b1 + k);
        v2f f2 = *(const v2f*)(b2 + k);
        v2f f3 = *(const v2f*)(b3 + k);

        acc0a = __builtin_amdgcn_wmma_f32_16x16x4_f32(false, aA, false, f0,
                                                      (short)0, acc0a, false, false);
        acc1a = __builtin_amdgcn_wmma_f32_16x16x4_f32(false, aA, false, f1,
                                                      (short)0, acc1a, false, false);
        acc2a = __builtin_amdgcn_wmma_f32_16x16x4_f32(false, aA, false, f2,
                                                      (short)0, acc2a, false, false);
        acc3a = __builtin_amdgcn_wmma_f32_16x16x4_f32(false, aA, false, f3,
                                                      (short)0, acc3a, false, false);
        acc0b = __builtin_amdgcn_wmma_f32_16x16x4_f32(false, aB, false, f0,
                                                      (short)0, acc0b, false, false);
        acc1b = __builtin_amdgcn_wmma_f32_16x16x4_f32(false, aB, false, f1,
                                                      (short)0, acc1b, false, false);
        acc2b = __builtin_amdgcn_wmma_f32_16x16x4_f32(false, aB, false, f2,
                                                      (short)0, acc2b, false, false);
        acc3b = __builtin_amdgcn_wmma_f32_16x16x4_f32(false, aB, false, f3,
                                                      (short)0, acc3b, false, false);
    }

    // C/D layout: VGPR v -> row (tileRowBase + v + 8*half), col (tile*16 + idx)
    const int row0 = rowBase +  0 + 8 * half;
    const int row1 = rowBase + 16 + 8 * half;
#pragma unroll
    for (int v = 0; v < 8; ++v) {
        size_t ra = (size_t)(row0 + v) * NEXP;
        logits[ra +  0 + idx] = acc0a[v];
        logits[ra + 16 + idx] = acc1a[v];
        logits[ra + 32 + idx] = acc2a[v];
        logits[ra + 48 + idx] = acc3a[v];
        size_t rb = (size_t)(row1 + v) * NEXP;
        logits[rb +  0 + idx] = acc0b[v];
        logits[rb + 16 + idx] = acc1b[v];
        logits[rb + 32 + idx] = acc2b[v];
        logits[rb + 48 + idx] = acc3b[v];
    }
}

// ---------------------------------------------------------------------------
// Softmax + top-8 + renormalize. One thread (lane) per row; 64 logits kept in
// registers (all loops fully unrolled -> no scratch). Renormalized top-k
// probs == softmax over the 8 selected logits, so the full 64-wide softmax
// is never materialized. Strict '>' with ascending scan = first-index-wins
// tie order, matching jax.lax.top_k.
// ---------------------------------------------------------------------------
__global__ __launch_bounds__(256)
void router_topk(const float* __restrict__ logits,
                 float* __restrict__ topv,
                 int* __restrict__ topi,
                 int T)
{
    int row = blockIdx.x * blockDim.x + threadIdx.x;
    if (row >= T) return;

    float v[NEXP];
    const v4f* p = (const v4f*)(logits + (size_t)row * NEXP);
#pragma unroll
    for (int j = 0; j < NEXP / 4; ++j) {
        v4f q = p[j];
        v[4 * j + 0] = q.x;
        v[4 * j + 1] = q.y;
        v[4 * j + 2] = q.z;
        v[4 * j + 3] = q.w;
    }

    const float NEG = -__builtin_inff();
    float selv[TOPK];
    int   seli[TOPK];

#pragma unroll
    for (int s = 0; s < TOPK; ++s) {
        float best = NEG;
        int   bi   = 0;
#pragma unroll
        for (int j = 0; j < NEXP; ++j) {
            if (v[j] > best) { best = v[j]; bi = j; }
        }
        selv[s] = best;
        seli[s] = bi;
#pragma unroll
        for (int j = 0; j < NEXP; ++j) {
            if (j == bi) v[j] = NEG;
        }
    }

    // softmax over the 8 selected logits (== renormalized top-k probs)
    float m = selv[0];            // first selection is the global max
    float e[TOPK];
    float sum = 0.0f;
#pragma unroll
    for (int s = 0; s < TOPK; ++s) { e[s] = __expf(selv[s] - m); sum += e[s]; }
    float inv = 1.0f / sum;

    float* ov = topv + (size_t)row * TOPK;
    int*   oi = topi + (size_t)row * TOPK;
#pragma unroll
    for (int s = 0; s < TOPK; ++s) { ov[s] = e[s] * inv; oi[s] = seli[s]; }
}

// ---------------------------------------------------------------------------
extern "C" void kernel_launch(void* const* d_in, const int* in_sizes, int n_in,
                              void* d_out, int out_size, void* d_ws, size_t ws_size,
                              hipStream_t stream)
{
    const float* hs = (const float*)d_in[0];   // [T, 2048] f32
    const float* wt = (const float*)d_in[1];   // [64, 2048] f32

    const int T = in_sizes[0] / HDIM;          // 16384

    // d_out flat layout (return order): logits f32 [T*64],
    // top_values f32 [T*8], indices i32 [T*8]
    float* logits = (float*)d_out;
    float* topv   = logits + (size_t)T * NEXP;
    int*   topi   = (int*)(topv + (size_t)T * TOPK);

    router_gemm_wmma<<<T / ROWS_PER_BLOCK, 128, 0, stream>>>(hs, wt, logits);
    router_topk<<<(T + 255) / 256, 256, 0, stream>>>(logits, topv, topi, T);
}